// MultiBoxLoss_51591147160002
// MI455X (gfx1250) — compile-verified
//
#include <hip/hip_runtime.h>
#include <hip/hip_bf16.h>

// ---------------- problem constants (fixed by reference setup_inputs) -------
constexpr int B_ = 32;        // batch
constexpr int P_ = 32768;     // priors
constexpr int O_ = 64;        // objects (truths) per image
constexpr int C_ = 81;        // classes
constexpr float THR = 0.5f;
constexpr int   NEGPOS = 3;
constexpr float VAR0 = 0.1f, VAR1 = 0.2f;

// ---------------- workspace layout (all offsets 256B aligned) ---------------
constexpr size_t OFF_GKEY   = 0;                          // B*O u64      = 16384
constexpr size_t OFF_NUMPOS = 16384;                      // B int        = 128
constexpr size_t OFF_LCNEG  = 16384 + 256;                // B float      = 128
constexpr size_t OFF_PARTL  = 16384 + 512;                // 4096 float   = 16384
constexpr size_t OFF_PARTC  = OFF_PARTL + 16384;          // 65536 float  = 262144
constexpr size_t OFF_BTO    = OFF_PARTC + 262144;         // B*P float    = 4 MB
constexpr size_t OFF_MIN    = OFF_BTO + (size_t)B_*P_*4;  // B*P float    = 4 MB
constexpr size_t OFF_BTI    = OFF_MIN + (size_t)B_*P_*4;  // B*P u8       = 1 MB
constexpr size_t OFF_CONF   = OFF_BTI + (size_t)B_*P_;    // B*P u8       = 1 MB

typedef float v2f __attribute__((ext_vector_type(2)));
typedef float v8f __attribute__((ext_vector_type(8)));

// ============ Kernel A: jaccard matching, per-prior best truth + per-truth
// best prior (packed u64 atomic max: [overlap_bits<<32 | ~p] -> tie = low p) ==
__global__ void kA_match(const float* __restrict__ priors,
                         const float* __restrict__ targets,
                         float* __restrict__ bto,
                         unsigned char* __restrict__ bti,
                         unsigned long long* __restrict__ gkey) {
  __shared__ float tb[O_ * 5];
  __shared__ unsigned long long lkey[O_];
  const int tid = threadIdx.x;
  const int blocksPerBatch = P_ / 256;
  const int b = blockIdx.x / blocksPerBatch;
  const int p = (blockIdx.x % blocksPerBatch) * 256 + tid;
  for (int i = tid; i < O_ * 5; i += 256) tb[i] = targets[(size_t)b * O_ * 5 + i];
  for (int i = tid; i < O_; i += 256) lkey[i] = 0ull;
  __syncthreads();

  const float cx = priors[p * 4 + 0], cy = priors[p * 4 + 1];
  const float w  = priors[p * 4 + 2], h  = priors[p * 4 + 3];
  const float px0 = cx - 0.5f * w, py0 = cy - 0.5f * h;
  const float px1 = cx + 0.5f * w, py1 = cy + 0.5f * h;
  const float areaP = w * h;

  float best = -1.0f; int bestIdx = 0;
  for (int o = 0; o < O_; ++o) {
    const float t0 = tb[o*5+0], t1 = tb[o*5+1], t2 = tb[o*5+2], t3 = tb[o*5+3];
    float iw = fminf(t2, px1) - fmaxf(t0, px0); iw = fmaxf(iw, 0.0f);
    float ih = fminf(t3, py1) - fmaxf(t1, py0); ih = fmaxf(ih, 0.0f);
    const float inter = iw * ih;
    const float areaT = (t2 - t0) * (t3 - t1);
    const float ov = inter / (areaT + areaP - inter);
    if (ov > best) { best = ov; bestIdx = o; }   // strict > keeps first idx (argmax tie rule)
    const unsigned long long key =
        ((unsigned long long)__float_as_uint(ov) << 32) |
        (unsigned long long)(0xFFFFFFFFu - (unsigned)p);
    atomicMax(&lkey[o], key);
  }
  bto[(size_t)b * P_ + p] = best;
  bti[(size_t)b * P_ + p] = (unsigned char)bestIdx;
  __syncthreads();
  for (int o = tid; o < O_; o += 256)
    atomicMax(&gkey[b * O_ + o], lkey[o]);
}

// ===== Kernel B: sequential override (last-wins on duplicates, ascending o) ==
__global__ void kB_override(const unsigned long long* __restrict__ gkey,
                            float* __restrict__ bto,
                            unsigned char* __restrict__ bti) {
  const int b = threadIdx.x;
  if (b >= B_) return;
  for (int o = 0; o < O_; ++o) {
    const unsigned long long key = gkey[b * O_ + o];
    const unsigned int p = 0xFFFFFFFFu - (unsigned int)(key & 0xFFFFFFFFull);
    bto[(size_t)b * P_ + p] = 2.0f;
    bti[(size_t)b * P_ + p] = (unsigned char)o;
  }
}

// ===== Kernel C: conf_t, box encode, smooth-L1 partials, num_pos ============
__global__ void kC_encode(const float* __restrict__ loc_data,
                          const float* __restrict__ priors,
                          const float* __restrict__ targets,
                          const float* __restrict__ bto,
                          const unsigned char* __restrict__ bti,
                          unsigned char* __restrict__ conf_u8,
                          float* __restrict__ partialsL,
                          int* __restrict__ num_pos) {
  __shared__ float tb[O_ * 5];
  __shared__ float redf[256];
  __shared__ int   redi[256];
  const int tid = threadIdx.x;
  const int blocksPerBatch = P_ / 256;
  const int b = blockIdx.x / blocksPerBatch;
  const int p = (blockIdx.x % blocksPerBatch) * 256 + tid;
  for (int i = tid; i < O_ * 5; i += 256) tb[i] = targets[(size_t)b * O_ * 5 + i];
  __syncthreads();

  const size_t gi = (size_t)b * P_ + p;
  const float ov = bto[gi];
  const int o = bti[gi];
  int conf = (int)tb[o * 5 + 4] + 1;
  if (ov < THR) conf = 0;
  conf_u8[gi] = (unsigned char)conf;

  float lsum = 0.0f;
  const int posi = (conf > 0) ? 1 : 0;
  if (posi) {
    const float m0 = tb[o*5+0], m1 = tb[o*5+1], m2 = tb[o*5+2], m3 = tb[o*5+3];
    const float cx = priors[p*4+0], cy = priors[p*4+1];
    const float w  = priors[p*4+2], h  = priors[p*4+3];
    float lt[4];
    lt[0] = ((m0 + m2) * 0.5f - cx) / (VAR0 * w);
    lt[1] = ((m1 + m3) * 0.5f - cy) / (VAR0 * h);
    lt[2] = logf((m2 - m0) / w) / VAR1;
    lt[3] = logf((m3 - m1) / h) / VAR1;
    for (int k = 0; k < 4; ++k) {
      const float d = loc_data[gi * 4 + k] - lt[k];
      const float ad = fabsf(d);
      lsum += (ad < 1.0f) ? 0.5f * d * d : ad - 0.5f;
    }
  }
  redf[tid] = lsum; redi[tid] = posi;
  __syncthreads();
  for (int s = 128; s > 0; s >>= 1) {
    if (tid < s) { redf[tid] += redf[tid + s]; redi[tid] += redi[tid + s]; }
    __syncthreads();
  }
  if (tid == 0) {
    partialsL[blockIdx.x] = redf[0];
    atomicAdd(&num_pos[b], redi[0]);
  }
}

// ===== Kernel D: logsumexp over 81 classes, 16 rows per 1-wave block.
// Row sums of exp() done as D = E x ones via V_WMMA_F32_16X16X4_F32. =========
__global__ void kD_conf(const float* __restrict__ conf_data,
                        const unsigned char* __restrict__ conf_u8,
                        float* __restrict__ mining,
                        float* __restrict__ partialsC) {
  __shared__ float orig[16 * 81];
  __shared__ float exps[16 * 84];   // padded to a multiple of 4 K-chunks
  __shared__ float rmax[16];
  __shared__ float rsum[16];
  const int lane = threadIdx.x;                 // wave32, one wave per block
  const size_t r0 = (size_t)blockIdx.x * 16;    // first global row of tile
  const size_t base = r0 * C_;

  for (int i = lane; i < 16 * 81; i += 32) orig[i] = conf_data[base + i];
  __syncthreads();

  // stable max per row: 2 lanes per row
  const int row = lane >> 1, sub = lane & 1;
  float m = -3.402823466e38f;
  for (int c = sub; c < C_; c += 2) m = fmaxf(m, orig[row * 81 + c]);
  m = fmaxf(m, __shfl_xor(m, 1));
  if (sub == 0) rmax[row] = m;
  for (int c = sub; c < C_; c += 2) exps[row * 84 + c] = expf(orig[row * 81 + c] - m);
  if (sub == 0) { exps[row*84+81] = 0.0f; exps[row*84+82] = 0.0f; exps[row*84+83] = 0.0f; }
  __syncthreads();

  // WMMA row-sum: A = 16x4 chunk of exps, B = ones (layout-agnostic), 21 chunks cover 84 K.
  // A layout: lanes 0-15 hold K0,K1 for M=lane; lanes 16-31 hold K2,K3 for M=lane-16.
  v8f acc = {};
  v2f bones; bones.x = 1.0f; bones.y = 1.0f;
  const int arow = lane & 15;
  const int koff = (lane >> 4) << 1;
  for (int k = 0; k < 21; ++k) {
    v2f a;
    a.x = exps[arow * 84 + 4 * k + koff];
    a.y = exps[arow * 84 + 4 * k + koff + 1];
    acc = __builtin_amdgcn_wmma_f32_16x16x4_f32(false, a, false, bones,
                                                (short)0, acc, false, false);
  }
  // D layout: VGPR r -> M=r (lanes 0-15) / M=r+8 (lanes 16-31); value replicated over N.
  if (lane == 0)  for (int i = 0; i < 8; ++i) rsum[i]     = acc[i];
  if (lane == 16) for (int i = 0; i < 8; ++i) rsum[8 + i] = acc[i];
  __syncthreads();

  float posLoss = 0.0f;
  if (lane < 16) {
    const int mr = lane;
    const float lse = rmax[mr] + logf(rsum[mr]);
    const int ct = conf_u8[r0 + mr];
    const float loss = lse - orig[mr * 81 + ct];
    const bool pos = ct > 0;
    mining[r0 + mr] = pos ? 0.0f : loss;
    posLoss = pos ? loss : 0.0f;
  }
  for (int s = 16; s > 0; s >>= 1) posLoss += __shfl_xor(posLoss, s);
  if (lane == 0) partialsC[blockIdx.x] = posLoss;
}

// ===== Kernel E: exact top-K sum per batch via MSB-first radix select ========
// mining >= 0, so uint bit order == float order. Exact under ties:
// sum_topK = sum(values > v) + (K - count_gt) * v, v = K-th largest.
__global__ void kE_topk(const float* __restrict__ mining,
                        const int* __restrict__ num_pos,
                        float* __restrict__ lossCneg) {
  __shared__ unsigned int counts[256];
  __shared__ unsigned int sPrefix, sMask, sRemaining;
  __shared__ float redf[256];
  __shared__ unsigned int redu[256];
  const int tid = threadIdx.x;
  const int b = blockIdx.x;
  const float* row = mining + (size_t)b * P_;

  long long K = (long long)NEGPOS * (long long)num_pos[b];
  if (K > P_ - 1) K = P_ - 1;
  if (K <= 0) { if (tid == 0) lossCneg[b] = 0.0f; return; }  // uniform per block

  unsigned int prefix = 0, mask = 0, remaining = (unsigned int)K;
  for (int pass = 3; pass >= 0; --pass) {
    const int shift = pass * 8;
    counts[tid] = 0u;
    __syncthreads();
    for (int i = tid; i < P_; i += 256) {
      const unsigned int u = __float_as_uint(row[i]);
      if ((u & mask) == prefix) atomicAdd(&counts[(u >> shift) & 255u], 1u);
    }
    __syncthreads();
    if (tid == 0) {
      unsigned int rem = remaining;
      int d = 255;
      for (; d > 0; --d) {
        const unsigned int c = counts[d];
        if (rem <= c) break;
        rem -= c;
      }
      sPrefix = prefix | ((unsigned int)d << shift);
      sMask = mask | (255u << shift);
      sRemaining = rem;
    }
    __syncthreads();
    prefix = sPrefix; mask = sMask; remaining = sRemaining;
    __syncthreads();
  }

  const unsigned int vbits = prefix;
  const float v = __uint_as_float(vbits);
  float lsum = 0.0f; unsigned int lcnt = 0;
  for (int i = tid; i < P_; i += 256) {
    const unsigned int u = __float_as_uint(row[i]);
    if (u > vbits) { lsum += row[i]; lcnt++; }
  }
  redf[tid] = lsum; redu[tid] = lcnt;
  __syncthreads();
  for (int s = 128; s > 0; s >>= 1) {
    if (tid < s) { redf[tid] += redf[tid + s]; redu[tid] += redu[tid + s]; }
    __syncthreads();
  }
  if (tid == 0)
    lossCneg[b] = redf[0] + (float)(K - (long long)redu[0]) * v;
}

// ===== Kernel F: deterministic final reductions -> d_out[0..1] ==============
__global__ void kF_final(const float* __restrict__ partialsL, int nL,
                         const float* __restrict__ partialsC, int nC,
                         const float* __restrict__ lossCneg,
                         const int* __restrict__ num_pos,
                         float* __restrict__ out) {
  __shared__ float red[256];
  __shared__ int redi[256];
  const int tid = threadIdx.x;

  float sL = 0.0f;
  for (int i = tid; i < nL; i += 256) sL += partialsL[i];
  red[tid] = sL; __syncthreads();
  for (int s = 128; s > 0; s >>= 1) { if (tid < s) red[tid] += red[tid + s]; __syncthreads(); }
  const float sumL = red[0]; __syncthreads();

  float sC = 0.0f;
  for (int i = tid; i < nC; i += 256) sC += partialsC[i];
  red[tid] = sC; __syncthreads();
  for (int s = 128; s > 0; s >>= 1) { if (tid < s) red[tid] += red[tid + s]; __syncthreads(); }
  const float sumCpos = red[0]; __syncthreads();

  float sN = 0.0f;
  for (int i = tid; i < B_; i += 256) sN += lossCneg[i];
  red[tid] = sN; __syncthreads();
  for (int s = 128; s > 0; s >>= 1) { if (tid < s) red[tid] += red[tid + s]; __syncthreads(); }
  const float sumCneg = red[0]; __syncthreads();

  int tp = 0;
  for (int i = tid; i < B_; i += 256) tp += num_pos[i];
  redi[tid] = tp; __syncthreads();
  for (int s = 128; s > 0; s >>= 1) { if (tid < s) redi[tid] += redi[tid + s]; __syncthreads(); }

  if (tid == 0) {
    const float N = fmaxf((float)redi[0], 1.0f);
    out[0] = sumL / N;
    out[1] = (sumCpos + sumCneg) / N;
  }
}

// ============================ host launcher =================================
extern "C" void kernel_launch(void* const* d_in, const int* in_sizes, int n_in,
                              void* d_out, int out_size, void* d_ws, size_t ws_size,
                              hipStream_t stream) {
  (void)in_sizes; (void)n_in; (void)out_size; (void)ws_size;
  const float* loc_data  = (const float*)d_in[0];   // (B,P,4)
  const float* conf_data = (const float*)d_in[1];   // (B,P,81)
  const float* priors    = (const float*)d_in[2];   // (P,4)
  const float* targets   = (const float*)d_in[3];   // (B,O,5)
  float* out = (float*)d_out;

  unsigned char* ws = (unsigned char*)d_ws;
  unsigned long long* gkey = (unsigned long long*)(ws + OFF_GKEY);
  int*   num_pos   = (int*)  (ws + OFF_NUMPOS);
  float* lossCneg  = (float*)(ws + OFF_LCNEG);
  float* partialsL = (float*)(ws + OFF_PARTL);
  float* partialsC = (float*)(ws + OFF_PARTC);
  float* bto       = (float*)(ws + OFF_BTO);
  float* mining    = (float*)(ws + OFF_MIN);
  unsigned char* bti     = ws + OFF_BTI;
  unsigned char* conf_u8 = ws + OFF_CONF;

  // atomically-accumulated buffers must start zeroed every call
  hipMemsetAsync(gkey, 0, (size_t)B_ * O_ * sizeof(unsigned long long), stream);
  hipMemsetAsync(num_pos, 0, (size_t)B_ * sizeof(int), stream);

  const int blocksBP = B_ * (P_ / 256);                  // 4096
  kA_match<<<blocksBP, 256, 0, stream>>>(priors, targets, bto, bti, gkey);
  kB_override<<<1, 32, 0, stream>>>(gkey, bto, bti);
  kC_encode<<<blocksBP, 256, 0, stream>>>(loc_data, priors, targets, bto, bti,
                                          conf_u8, partialsL, num_pos);
  const int blocksD = (B_ * P_) / 16;                    // 65536, one wave each
  kD_conf<<<blocksD, 32, 0, stream>>>(conf_data, conf_u8, mining, partialsC);
  kE_topk<<<B_, 256, 0, stream>>>(mining, num_pos, lossCneg);
  kF_final<<<1, 256, 0, stream>>>(partialsL, blocksBP, partialsC, blocksD,
                                  lossCneg, num_pos, out);
}